// AdaptiveInput_120259084974
// MI455X (gfx1250) — compile-verified
//
#include <hip/hip_runtime.h>
#include <hip/hip_bf16.h>

typedef float v2f __attribute__((ext_vector_type(2)));
typedef float v8f __attribute__((ext_vector_type(8)));

#define NTOK_MAX 32768
#define D_OUT 1024
#define KC 64          // K chunk staged in LDS
#define LDS_STRIDE 66  // padded row stride (floats) to avoid bank conflicts

__global__ void ai_init_counts(int* counts) {
    if (threadIdx.x < 4) counts[threadIdx.x] = 0;
}

__global__ void ai_bucket(const int* __restrict__ input, int N,
                          int* __restrict__ counts,
                          int* __restrict__ ptok, int* __restrict__ ploc) {
    int n = blockIdx.x * blockDim.x + threadIdx.x;
    if (n >= N) return;
    int id = input[n];
    int c, lo;
    if (id < 10000)       { c = 0; lo = 0; }
    else if (id < 60000)  { c = 1; lo = 10000; }
    else if (id < 190000) { c = 2; lo = 60000; }
    else                  { c = 3; lo = 190000; }
    int pos = atomicAdd(&counts[c], 1);
    ptok[(size_t)c * N + pos] = n;
    ploc[(size_t)c * N + pos] = id - lo;
}

// grid: (D_OUT/128, ceil(N/16), 4), block: 256 threads = 8 waves.
// Each wave computes one 16x16 f32 tile via V_WMMA_F32_16X16X4_F32.
__global__ void __launch_bounds__(256)
ai_gemm(const float* __restrict__ e0, const float* __restrict__ w0,
        const float* __restrict__ e1, const float* __restrict__ w1,
        const float* __restrict__ e2, const float* __restrict__ w2,
        const float* __restrict__ e3, const float* __restrict__ w3,
        const int* __restrict__ counts,
        const int* __restrict__ ptok, const int* __restrict__ ploc,
        float* __restrict__ out, int N) {
    __shared__ __align__(16) float As[16 * LDS_STRIDE];
    __shared__ int tokS[16];
    __shared__ int locS[16];

    const int c = blockIdx.z;
    const float* emb; const float* w; int h;
    switch (c) {
        case 0:  emb = e0; w = w0; h = 1024; break;
        case 1:  emb = e1; w = w1; h = 256;  break;
        case 2:  emb = e2; w = w2; h = 64;   break;
        default: emb = e3; w = w3; h = 16;   break;
    }

    const int cnt  = counts[c];
    const int row0 = blockIdx.y * 16;
    if (row0 >= cnt) return;  // block-uniform: EXEC stays all-ones below

    const int tid  = threadIdx.x;
    const int lane = tid & 31;
    const int wave = tid >> 5;
    const int cb   = blockIdx.x * 128 + wave * 16;  // output column base
    const int ln   = lane & 15;                     // N (or M) index within tile
    const int hi2  = (lane >> 4) * 2;               // K sub-offset per f32 WMMA layout

    if (tid < 16) {
        int r = row0 + tid;
        tokS[tid] = (r < cnt) ? ptok[(size_t)c * N + r] : 0;
        locS[tid] = (r < cnt) ? ploc[(size_t)c * N + r] : 0;
    }
    __syncthreads();

    v8f acc = {};
    for (int k0 = 0; k0 < h; k0 += KC) {
        const int kc  = (h - k0 >= KC) ? KC : (h - k0);  // 64 or 16
        const int kcs = (kc == KC) ? 6 : 4;

        // Cooperative gather of A chunk (16 embedding rows x kc) into LDS.
        for (int e = tid; e < 16 * kc; e += 256) {
            int row = e >> kcs;
            int col = e & (kc - 1);
            As[row * LDS_STRIDE + col] = emb[(size_t)locS[row] * h + k0 + col];
        }
        __syncthreads();

        const float* wp = w + (size_t)(cb + ln) * h + k0 + hi2;  // B: W[col][k]
        const float* ap = &As[ln * LDS_STRIDE + hi2];
        const int nsteps = kc >> 2;
        for (int s = 0; s < nsteps; ++s) {
            v2f a = *(const v2f*)(ap + s * 4);   // ds_load_b64
            v2f b = *(const v2f*)(wp + s * 4);   // global_load_b64 (L2-resident)
            acc = __builtin_amdgcn_wmma_f32_16x16x4_f32(
                false, a, false, b, (short)0, acc, false, false);
        }
        __syncthreads();
    }

    // Store: C VGPR v holds row (v + 8*(lane>=16)), col = lane&15.
    const int mval = (cnt - row0 < 16) ? (cnt - row0) : 16;
    const int mb   = (lane >> 4) * 8;
#pragma unroll
    for (int v = 0; v < 8; ++v) {
        int m = mb + v;
        if (m < mval)
            out[(size_t)tokS[m] * D_OUT + cb + ln] = acc[v];
    }
}

extern "C" void kernel_launch(void* const* d_in, const int* in_sizes, int n_in,
                              void* d_out, int out_size, void* d_ws, size_t ws_size,
                              hipStream_t stream) {
    const int*   input = (const int*)d_in[0];
    const float* e0 = (const float*)d_in[1];
    const float* w0 = (const float*)d_in[2];
    const float* e1 = (const float*)d_in[3];
    const float* w1 = (const float*)d_in[4];
    const float* e2 = (const float*)d_in[5];
    const float* w2 = (const float*)d_in[6];
    const float* e3 = (const float*)d_in[7];
    const float* w3 = (const float*)d_in[8];
    float* out = (float*)d_out;
    const int N = in_sizes[0];

    // Workspace layout: [counts(4 ints), pad to 256B][ptok 4*N ints][ploc 4*N ints]
    int* counts = (int*)d_ws;
    int* ptok   = (int*)((char*)d_ws + 256);
    int* ploc   = ptok + 4 * (size_t)N;

    ai_init_counts<<<1, 32, 0, stream>>>(counts);
    ai_bucket<<<(N + 255) / 256, 256, 0, stream>>>(input, N, counts, ptok, ploc);

    dim3 grid(D_OUT / 128, (N + 15) / 16, 4);
    ai_gemm<<<grid, 256, 0, stream>>>(e0, w0, e1, w1, e2, w2, e3, w3,
                                      counts, ptok, ploc, out, N);
}